// PureCrf_11690900979887
// MI455X (gfx1250) — compile-verified
//
#include <hip/hip_runtime.h>
#include <hip/hip_bf16.h>
#include <stdint.h>

#define Tn 512
#define Bn 64
#define Fn 32
#define Kn 32

typedef __attribute__((ext_vector_type(2))) float v2f;
typedef __attribute__((ext_vector_type(8))) float v8f;
typedef __attribute__((ext_vector_type(4))) unsigned int u32x4;
typedef __attribute__((ext_vector_type(8))) unsigned int u32x8;

// ---------------------------------------------------------------------------
// Kernel 1: emission features.  One wave per (t,b); lane = tag k.
// feats[t,b,k] = sum_f W[words[t,b,f] + k]   (valid steps)
//             = 32 * W[0]                    (padding steps, per reference mask)
// The 32 lookups per (t,b,f) are contiguous -> coalesced 128B loads, L2-resident.
// ---------------------------------------------------------------------------
__global__ __launch_bounds__(256) void crf_emit(const int* __restrict__ words,
                                                const int* __restrict__ len,
                                                const float* __restrict__ W,
                                                float* __restrict__ feats) {
  const int lane = threadIdx.x;                 // k (and f for the staging load)
  const int tb   = blockIdx.x * 8 + threadIdx.y;
  const int t    = tb >> 6;                     // / 64
  const int b    = tb & 63;
  const bool valid = t < len[b];                // uniform per wave
  float acc;
  if (valid) {
    const int wf = words[(size_t)tb * Fn + lane];   // lane holds f-th word id
    acc = 0.0f;
#pragma unroll
    for (int f = 0; f < Fn; ++f) {
      const int w = __shfl(wf, f, 32);          // broadcast word id f
      acc += W[w + lane];                       // contiguous 32-float gather
    }
  } else {
    acc = 32.0f * W[0];
  }
  feats[(size_t)tb * Kn + lane] = acc;
}

// ---------------------------------------------------------------------------
// TDM: async 1-D tile load (2 KB of feats) into LDS.  D# per CDNA5 ISA ch.8.
// ---------------------------------------------------------------------------
__device__ __forceinline__ void tdm_load_feats(uint32_t lds_off, const float* gptr) {
  const uint64_t ga = (uint64_t)(uintptr_t)gptr;
  u32x4 g0;
  g0.x = (unsigned)__builtin_amdgcn_readfirstlane((int)1u);            // count=1
  g0.y = (unsigned)__builtin_amdgcn_readfirstlane((int)lds_off);       // lds_addr
  g0.z = (unsigned)__builtin_amdgcn_readfirstlane((int)(uint32_t)ga);  // ga[31:0]
  g0.w = (unsigned)__builtin_amdgcn_readfirstlane(
      (int)((((uint32_t)(ga >> 32)) & 0x01FFFFFFu) | 0x80000000u));    // ga[56:32] | type=2
  u32x8 g1;
  g1.s0 = 2u << 16;            // workgroup_mask=0, data_size=2 (4B)
  g1.s1 = (512u & 0xFFFFu) << 16;        // tensor_dim0[15:0] @ bits 63:48
  g1.s2 = (512u >> 16) | (1u << 16);     // tensor_dim0[31:16], tensor_dim1=1
  g1.s3 = (512u << 16);                  // tensor_dim1[31:16]=0, tile_dim0=512
  g1.s4 = 0u;                            // tile_dim1=0 (unused), tile_dim2=0
  g1.s5 = 512u;                          // tensor_dim0_stride[31:0]
  g1.s6 = 0u;
  g1.s7 = 0u;
  asm volatile("tensor_load_to_lds %0, %1" :: "s"(g0), "s"(g1) : "memory");
}

// ---------------------------------------------------------------------------
// Kernel 2: sequential CRF scan.  4 blocks x 256 threads; block owns 16 batches.
// Forward logsumexp via chained V_WMMA_F32_16X16X4_F32 (waves 0-1, 8 K-steps);
// Viterbi max/argmax on VALU; feats double-buffered through the TDM.
// ---------------------------------------------------------------------------
__global__ __launch_bounds__(256) void crf_scan(const float* __restrict__ trans_g,
                                                const int* __restrict__ len,
                                                const float* __restrict__ feats,
                                                float* __restrict__ fwd,
                                                float* __restrict__ max_s,
                                                float* __restrict__ max_idx) {
  __shared__ float alpha[16][33];
  __shared__ float delta[16][33];
  __shared__ float Pm[16][33];      // exp(alpha - rowmax)
  __shared__ float Em[32][33];      // exp(transitions)
  __shared__ float mrow[16];
  __shared__ int   slen[16];
  __shared__ float ftile[2][16][32];

  const int tid   = threadIdx.x;
  const int lane  = tid & 31;
  const int wave  = tid >> 5;       // 0..7 ; also the low batch row this thread owns
  const int bbase = blockIdx.x * 16;

  // --- preamble: E = exp(trans), lengths, register-cached Viterbi column ----
  for (int e = tid; e < Kn * Kn; e += 256) {
    Em[e >> 5][e & 31] = __expf(trans_g[e]);
  }
  if (tid < 16) slen[tid] = len[bbase + tid];

  float tc[32];                     // trans[i][vj] column, vj = lane
#pragma unroll
  for (int i = 0; i < 32; ++i) tc[i] = trans_g[i * Kn + lane];

  const uint32_t ldsb = (uint32_t)(uintptr_t)(&ftile[0][0][0]);
  if (wave == 0) {                  // TDM: stage feats[0] and feats[1]
    tdm_load_feats(ldsb,        feats + (size_t)(0 * Bn + bbase) * Kn);
    tdm_load_feats(ldsb + 2048, feats + (size_t)(1 * Bn + bbase) * Kn);
    __builtin_amdgcn_s_wait_tensorcnt(0);
  }
  __syncthreads();

  // --- t = 0 init: alpha = delta = feats[0]; max_s[0]=feats[0]; idx[0]=0 ----
#pragma unroll
  for (int h = 0; h < 2; ++h) {
    const int e = tid + h * 256;
    const int b = e >> 5, j = e & 31;
    const float f0 = ftile[0][b][j];
    alpha[b][j] = f0;
    delta[b][j] = f0;
    max_s[(size_t)(bbase + b) * Kn + j]   = f0;
    max_idx[(size_t)(bbase + b) * Kn + j] = 0.0f;
  }
  __syncthreads();

  for (int t = 1; t < Tn; ++t) {
    const int cur = t & 1;
    if (wave == 0 && (t + 1) < Tn) {        // prefetch feats[t+1] into other buffer
      tdm_load_feats(ldsb + (uint32_t)(((t + 1) & 1) * 2048),
                     feats + ((size_t)(t + 1) * Bn + bbase) * Kn);
    }

    // ---- Phase A: row-max + P (forward prep), Viterbi compute + stores ----
    const int rA = wave, rB = wave + 8;     // the two batch rows this wave owns
    float a0 = alpha[rA][lane];
    float a1 = alpha[rB][lane];
    float m0 = a0, m1 = a1;
#pragma unroll
    for (int off = 16; off; off >>= 1) {
      m0 = fmaxf(m0, __shfl_xor(m0, off, 32));
      m1 = fmaxf(m1, __shfl_xor(m1, off, 32));
    }
    Pm[rA][lane] = __expf(a0 - m0);
    Pm[rB][lane] = __expf(a1 - m1);
    if (lane == 0) { mrow[rA] = m0; mrow[rB] = m1; }

    float best0 = -3.0e38f, best1 = -3.0e38f;
    int bp0 = 0, bp1 = 0;
#pragma unroll
    for (int i = 0; i < 32; ++i) {          // delta rows: LDS broadcast reads
      const float s0 = delta[rA][i] + tc[i];
      const float s1 = delta[rB][i] + tc[i];
      if (s0 > best0) { best0 = s0; bp0 = i; }
      if (s1 > best1) { best1 = s1; bp1 = i; }
    }
    const float f0 = ftile[cur][rA][lane];
    const float f1 = ftile[cur][rB][lane];
    const bool  v0 = t < slen[rA];
    const bool  v1 = t < slen[rB];
    const float nd0 = v0 ? best0 + f0 : delta[rA][lane];
    const float nd1 = v1 ? best1 + f1 : delta[rB][lane];
    const size_t ro = ((size_t)t * Bn + bbase) * Kn;
    max_s[ro + rA * Kn + lane]   = nd0;     // masked delta (reference semantics)
    max_s[ro + rB * Kn + lane]   = nd1;
    max_idx[ro + rA * Kn + lane] = (float)bp0;   // backpointers are UNmasked
    max_idx[ro + rB * Kn + lane] = (float)bp1;
    __syncthreads();

    // ---- Phase B: delta commit + forward WMMA -----------------------------
    delta[rA][lane] = nd0;
    delta[rB][lane] = nd1;

    if (wave < 2) {                         // two 16x16 output tiles (N-halves)
      const int half = lane >> 4, lm = lane & 15, n0 = wave * 16;
      v8f acc = {0.f, 0.f, 0.f, 0.f, 0.f, 0.f, 0.f, 0.f};
#pragma unroll
      for (int ks = 0; ks < 8; ++ks) {      // K = 32 in steps of 4
        const int k0 = 4 * ks + 2 * half;
        v2f av, bv;
        av.x = Pm[lm][k0];       av.y = Pm[lm][k0 + 1];     // A 16x4 layout
        bv.x = Em[k0][n0 + lm];  bv.y = Em[k0 + 1][n0 + lm];// B 4x16 layout
        acc = __builtin_amdgcn_wmma_f32_16x16x4_f32(
            false, av, false, bv, (short)0, acc, false, false);
      }
      float dv[8];
#pragma unroll
      for (int r = 0; r < 8; ++r) dv[r] = acc[r];
#pragma unroll
      for (int r = 0; r < 8; ++r) {         // D 16x16: vgpr r -> M=r (+8 hi half)
        const int m = r + (half << 3);
        const int j = n0 + lm;
        const float val = mrow[m] + __logf(dv[r]) + ftile[cur][m][j];
        if (t < slen[m]) alpha[m][j] = val; // freeze past sequence end
      }
    }
    if (wave == 0) __builtin_amdgcn_s_wait_tensorcnt(0);
    __syncthreads();
  }

  // --- final: forward_score[b] = logsumexp_j alpha[b,j] ---------------------
  if (tid < 16) {
    float mm = -3.0e38f;
#pragma unroll
    for (int j = 0; j < Kn; ++j) mm = fmaxf(mm, alpha[tid][j]);
    float s = 0.0f;
#pragma unroll
    for (int j = 0; j < Kn; ++j) s += __expf(alpha[tid][j] - mm);
    fwd[bbase + tid] = mm + __logf(s);
  }
}

// ---------------------------------------------------------------------------
extern "C" void kernel_launch(void* const* d_in, const int* in_sizes, int n_in,
                              void* d_out, int out_size, void* d_ws, size_t ws_size,
                              hipStream_t stream) {
  (void)in_sizes; (void)n_in; (void)out_size; (void)d_ws; (void)ws_size;
  const int*   words = (const int*)d_in[0];
  const int*   len   = (const int*)d_in[1];
  const float* W     = (const float*)d_in[2];
  const float* trans = (const float*)d_in[3];

  float* out = (float*)d_out;
  const size_t TBK = (size_t)Tn * Bn * Kn;
  float* fwd   = out;                    // [B]
  float* maxs  = out + Bn;               // [T,B,K]
  float* maxi  = out + Bn + TBK;         // [T,B,K] (int indices stored as float)
  float* feats = out + Bn + 2 * TBK;     // [T,B,K]

  dim3 eb(32, 8);
  crf_emit<<<(Tn * Bn) / 8, eb, 0, stream>>>(words, len, W, feats);
  crf_scan<<<Bn / 16, 256, 0, stream>>>(trans, len, feats, fwd, maxs, maxi);
}